// VGGTCrossFrameRKDAngleLoss_66176856097252
// MI455X (gfx1250) — compile-verified
//
#include <hip/hip_runtime.h>
#include <math.h>

#define B_   2
#define ST_  8
#define SS_  4
#define P_   1369
#define D_   1024
#define R_   128
#define NS_  128
#define K_   4
#define M_   (4 * P_)          // 5476 extra rows per batch
#define RK_  (R_ * K_)         // 512
#define EPS_UNIT 1e-8f
#define EPS_SIM  1e-12f

typedef float v2f __attribute__((ext_vector_type(2)));
typedef float v8f __attribute__((ext_vector_type(8)));

__device__ __forceinline__ float huber1(float d) {
  float ad = fabsf(d);
  return ad < 1.0f ? 0.5f * d * d : ad - 0.5f;
}

__global__ void k_init(float* acc) {
  if (threadIdx.x < 4) acc[threadIdx.x] = 0.0f;
}

// Gather ref rows (teacher frame 0 / student frame 0) + squared norms.
__global__ void k_gather_ref(const float* __restrict__ T, const float* __restrict__ S,
                             const int* __restrict__ refp,
                             float* __restrict__ ref_t, float* __restrict__ ref_s,
                             float* __restrict__ nr_t, float* __restrict__ nr_s) {
  __shared__ float red[256];
  int b = blockIdx.x / R_, r = blockIdx.x % R_;
  int p = refp[r];
  const float* st = T + (((long)b * ST_ + 0) * P_ + p) * D_;
  const float* ss = S + (((long)b * SS_ + 0) * P_ + p) * D_;
  float* dt = ref_t + ((long)b * R_ + r) * D_;
  float* ds = ref_s + ((long)b * R_ + r) * D_;
  float a = 0.f, c = 0.f;
  for (int d = threadIdx.x; d < D_; d += 256) {
    float vt = st[d], vs = ss[d];
    dt[d] = vt; ds[d] = vs;
    a += vt * vt; c += vs * vs;
  }
  red[threadIdx.x] = a; __syncthreads();
  for (int o = 128; o > 0; o >>= 1) { if (threadIdx.x < o) red[threadIdx.x] += red[threadIdx.x + o]; __syncthreads(); }
  if (threadIdx.x == 0) nr_t[b * R_ + r] = red[0];
  __syncthreads();
  red[threadIdx.x] = c; __syncthreads();
  for (int o = 128; o > 0; o >>= 1) { if (threadIdx.x < o) red[threadIdx.x] += red[threadIdx.x + o]; __syncthreads(); }
  if (threadIdx.x == 0) nr_s[b * R_ + r] = red[0];
}

// 1 / max(||extra row||, 1e-12) for each of the B*M extra rows (frames 1,3,5,7).
__global__ void k_inv_extra(const float* __restrict__ T, float* __restrict__ inv) {
  __shared__ float red[256];
  int b = blockIdx.x / M_;
  int m = blockIdx.x % M_;
  int fr = 1 + 2 * (m / P_);
  int pp = m % P_;
  const float* p = T + (((long)b * ST_ + fr) * P_ + pp) * D_;
  float a = 0.f;
  for (int d = threadIdx.x; d < D_; d += 256) { float v = p[d]; a += v * v; }
  red[threadIdx.x] = a; __syncthreads();
  for (int o = 128; o > 0; o >>= 1) { if (threadIdx.x < o) red[threadIdx.x] += red[threadIdx.x + o]; __syncthreads(); }
  if (threadIdx.x == 0) inv[blockIdx.x] = 1.0f / fmaxf(sqrtf(red[0]), EPS_SIM);
}

// C[bz][i][j] = sum_d A[bz][i][d] * Brow(bz,j)[d]   (row-dot GEMM, fp32 WMMA 16x16x4).
// mode==1: B row j lives in teacher "extra frame" layout; output scaled by colScale[bz*N+j].
// A 16x4 f32 layout: lane L holds M=L&15, K = k0 + 2*(L>>4) .. +1 (2 VGPRs).
// B 4x16 f32 layout: lane L holds N=L&15, K = k0 + 2*(L>>4) .. +1.
// C/D 16x16 f32: VGPR v -> M = v + 8*(L>>4), N = L&15.
__global__ void k_gemm_rowdot(const float* __restrict__ A,
                              const float* __restrict__ Bm,
                              float* __restrict__ C,
                              const float* __restrict__ colScale,
                              int Mrows, int Nrows, int Dk,
                              long strideA, long strideB, long strideC,
                              int mode) {
  int wid  = threadIdx.x >> 5;
  int lane = threadIdx.x & 31;
  int ntile = blockIdx.x * 4 + wid;
  int mtile = blockIdx.y;
  int bz    = blockIdx.z;
  int nTiles = (Nrows + 15) / 16;
  if (ntile >= nTiles) return;             // wave-uniform exit: EXEC stays all-1 for WMMA
  int half = lane >> 4;
  int l15  = lane & 15;
  int arow = mtile * 16 + l15; if (arow >= Mrows) arow = Mrows - 1;
  int brow = ntile * 16 + l15; if (brow >= Nrows) brow = Nrows - 1;
  const float* ap = A + (long)bz * strideA + (long)arow * Dk;
  long boff;
  if (mode == 1) {
    int fr = brow / P_;
    int pp = brow - fr * P_;
    boff = ((long)(1 + 2 * fr) * P_ + pp) * (long)D_;
  } else {
    boff = (long)brow * Dk;
  }
  const float* bp = Bm + (long)bz * strideB + boff;
  v8f acc = {};
  for (int k0 = 0; k0 < Dk; k0 += 4) {
    int ka = k0 + half * 2;
    v2f av, bv;
    av.x = ap[ka]; av.y = ap[ka + 1];
    bv.x = bp[ka]; bv.y = bp[ka + 1];
    acc = __builtin_amdgcn_wmma_f32_16x16x4_f32(false, av, false, bv,
                                                (short)0, acc, false, false);
  }
  int col = ntile * 16 + l15;
  float scale = 1.0f;
  if (mode == 1 && colScale && col < Nrows) scale = colScale[(long)bz * Nrows + col];
  for (int v = 0; v < 8; ++v) {
    int row = mtile * 16 + v + half * 8;
    if (row < Mrows && col < Nrows)
      C[(long)bz * strideC + (long)row * Nrows + col] = acc[v] * scale;
  }
}

// Top-4 per (b,r) row of sim: one wave per row.
__global__ void k_topk(const float* __restrict__ sim, int* __restrict__ idxOut) {
  __shared__ float sv[32 * K_];
  __shared__ int   si[32 * K_];
  int lane = threadIdx.x;
  const float* row = sim + (long)blockIdx.x * M_;
  float bv[K_] = {-3.0e38f, -3.0e38f, -3.0e38f, -3.0e38f};
  int   bi[K_] = {0x7fffffff, 0x7fffffff, 0x7fffffff, 0x7fffffff};
  for (int m = lane; m < M_; m += 32) {
    float v = row[m];
    if (v > bv[K_ - 1]) {
      bv[K_ - 1] = v; bi[K_ - 1] = m;
      for (int q = K_ - 1; q > 0; --q) {
        if (bv[q] > bv[q - 1]) {
          float tv = bv[q]; bv[q] = bv[q - 1]; bv[q - 1] = tv;
          int ti = bi[q]; bi[q] = bi[q - 1]; bi[q - 1] = ti;
        } else break;
      }
    }
  }
  for (int q = 0; q < K_; ++q) { sv[lane * K_ + q] = bv[q]; si[lane * K_ + q] = bi[q]; }
  __syncthreads();
  if (lane == 0) {
    float tv[K_] = {-3.0e38f, -3.0e38f, -3.0e38f, -3.0e38f};
    int   ti[K_] = {0x7fffffff, 0x7fffffff, 0x7fffffff, 0x7fffffff};
    for (int i = 0; i < 32 * K_; ++i) {
      float v = sv[i]; int id = si[i];
      if (v > tv[K_ - 1] || (v == tv[K_ - 1] && id < ti[K_ - 1])) {
        tv[K_ - 1] = v; ti[K_ - 1] = id;
        for (int q = K_ - 1; q > 0; --q) {
          if (tv[q] > tv[q - 1] || (tv[q] == tv[q - 1] && ti[q] < ti[q - 1])) {
            float a = tv[q]; tv[q] = tv[q - 1]; tv[q - 1] = a;
            int c = ti[q]; ti[q] = ti[q - 1]; ti[q - 1] = c;
          } else break;
        }
      }
    }
    for (int q = 0; q < K_; ++q) idxOut[blockIdx.x * K_ + q] = ti[q];
  }
}

// Per (b,r,k): gather sim_high, build b1_t/b1_s, and precompute scalar constants.
__global__ void k_brk(const float* __restrict__ T, const int* __restrict__ idx,
                      const float* __restrict__ ref_t, const float* __restrict__ ref_s,
                      float* __restrict__ simh, float* __restrict__ b1_t, float* __restrict__ b1_s,
                      float* __restrict__ nh,
                      float* __restrict__ c1t, float* __restrict__ c2t, float* __restrict__ c3t,
                      float* __restrict__ c1s, float* __restrict__ c2s, float* __restrict__ c3s) {
  __shared__ float red[256];
  __shared__ float tot[9];
  int brk = blockIdx.x;
  int b = brk / (R_ * K_);
  int r = (brk / K_) % R_;
  int m = idx[brk];
  int fr = 1 + 2 * (m / P_);
  int pp = m % P_;
  const float* hp = T + (((long)b * ST_ + fr) * P_ + pp) * D_;
  const float* rt = ref_t + ((long)b * R_ + r) * D_;
  const float* rs = ref_s + ((long)b * R_ + r) * D_;
  float h[4], dt[4], ds[4], rtv[4], rsv[4];
  float s[9] = {0, 0, 0, 0, 0, 0, 0, 0, 0};
  for (int i = 0; i < 4; ++i) {
    int d = threadIdx.x + i * 256;
    h[i] = hp[d]; rtv[i] = rt[d]; rsv[i] = rs[d];
    dt[i] = h[i] - rtv[i]; ds[i] = h[i] - rsv[i];
    s[0] += dt[i] * dt[i];  s[1] += ds[i] * ds[i];  s[2] += h[i] * h[i];
    s[3] += h[i] * dt[i];   s[4] += h[i] * ds[i];
    s[5] += rtv[i] * dt[i]; s[6] += rsv[i] * ds[i];
    s[7] += rtv[i] * h[i];  s[8] += rsv[i] * h[i];
  }
  for (int q = 0; q < 9; ++q) {
    red[threadIdx.x] = s[q]; __syncthreads();
    for (int o = 128; o > 0; o >>= 1) { if (threadIdx.x < o) red[threadIdx.x] += red[threadIdx.x + o]; __syncthreads(); }
    if (threadIdx.x == 0) tot[q] = red[0];
    __syncthreads();
  }
  float invT = 1.0f / fmaxf(sqrtf(tot[0]), EPS_UNIT);
  float invS = 1.0f / fmaxf(sqrtf(tot[1]), EPS_UNIT);
  float* oh  = simh + (long)brk * D_;
  float* obt = b1_t + (long)brk * D_;
  float* obs = b1_s + (long)brk * D_;
  for (int i = 0; i < 4; ++i) {
    int d = threadIdx.x + i * 256;
    oh[d]  = h[i];
    obt[d] = dt[i] * invT;
    obs[d] = ds[i] * invS;
  }
  if (threadIdx.x == 0) {
    nh[brk]  = tot[2];
    c2t[brk] = tot[3] * invT;  c2s[brk] = tot[4] * invS;   // dot(sim_high, b1)
    c1t[brk] = tot[5] * invT;  c1s[brk] = tot[6] * invS;   // dot(ref, b1)
    c3t[brk] = tot[7];         c3s[brk] = tot[8];          // dot(ref, sim_high)
  }
}

// Gather shared rows for one frame + squared norms.
__global__ void k_gather_sh(const float* __restrict__ T, const float* __restrict__ S,
                            const int* __restrict__ shp, int tf, int f,
                            float* __restrict__ sh_t, float* __restrict__ sh_s,
                            float* __restrict__ ns_t, float* __restrict__ ns_s) {
  __shared__ float red[256];
  int b = blockIdx.x / NS_, s = blockIdx.x % NS_;
  int p = shp[s];
  const float* st = T + (((long)b * ST_ + tf) * P_ + p) * D_;
  const float* ss = S + (((long)b * SS_ + f) * P_ + p) * D_;
  float* dt = sh_t + ((long)b * NS_ + s) * D_;
  float* ds = sh_s + ((long)b * NS_ + s) * D_;
  float a = 0.f, c = 0.f;
  for (int d = threadIdx.x; d < D_; d += 256) {
    float vt = st[d], vs = ss[d];
    dt[d] = vt; ds[d] = vs;
    a += vt * vt; c += vs * vs;
  }
  red[threadIdx.x] = a; __syncthreads();
  for (int o = 128; o > 0; o >>= 1) { if (threadIdx.x < o) red[threadIdx.x] += red[threadIdx.x + o]; __syncthreads(); }
  if (threadIdx.x == 0) ns_t[b * NS_ + s] = red[0];
  __syncthreads();
  red[threadIdx.x] = c; __syncthreads();
  for (int o = 128; o > 0; o >>= 1) { if (threadIdx.x < o) red[threadIdx.x] += red[threadIdx.x + o]; __syncthreads(); }
  if (threadIdx.x == 0) ns_s[b * NS_ + s] = red[0];
}

// Combine GEMM results into a1/a2/a3, huber, reduce, accumulate.
__global__ void k_epilogue(const float* __restrict__ G1t, const float* __restrict__ G1s,
                           const float* __restrict__ G2t, const float* __restrict__ G2s,
                           const float* __restrict__ G3t, const float* __restrict__ G3s,
                           const float* __restrict__ nst, const float* __restrict__ nss,
                           const float* __restrict__ nrt, const float* __restrict__ nrs,
                           const float* __restrict__ nh,
                           const float* __restrict__ c1t, const float* __restrict__ c2t, const float* __restrict__ c3t,
                           const float* __restrict__ c1s, const float* __restrict__ c2s, const float* __restrict__ c3s,
                           float* __restrict__ acc) {
  __shared__ float red[256];
  int gid = blockIdx.x * 256 + threadIdx.x;        // < B*R*NS*K = 131072
  int k = gid & (K_ - 1);
  int s = (gid >> 2) & (NS_ - 1);
  int r = (gid >> 9) & (R_ - 1);
  int b = gid >> 16;
  int rk = r * K_ + k;
  long bs = (long)b * NS_ + s;
  int br = b * R_ + r;
  int brk = br * K_ + k;
  float g1T = G1t[bs * RK_ + rk], g2T = G2t[bs * RK_ + rk], g3T = G3t[bs * R_ + r];
  float g1S = G1s[bs * RK_ + rk], g2S = G2s[bs * RK_ + rk], g3S = G3s[bs * R_ + r];
  float nsT = nst[bs], nsS = nss[bs];
  float nrT = nrt[br], nrS = nrs[br];
  float nhv = nh[brk];

  float i1T = 1.f / fmaxf(sqrtf(fmaxf(nsT - 2.f * g3T + nrT, 0.f)), EPS_UNIT);
  float i2T = 1.f / fmaxf(sqrtf(fmaxf(nsT - 2.f * g2T + nhv, 0.f)), EPS_UNIT);
  float a1T = (g1T - c1t[brk]) * i1T;
  float a2T = -(g1T - c2t[brk]) * i2T;
  float a3T = (nsT - g2T - g3T + c3t[brk]) * i1T * i2T;

  float i1S = 1.f / fmaxf(sqrtf(fmaxf(nsS - 2.f * g3S + nrS, 0.f)), EPS_UNIT);
  float i2S = 1.f / fmaxf(sqrtf(fmaxf(nsS - 2.f * g2S + nhv, 0.f)), EPS_UNIT);
  float a1S = (g1S - c1s[brk]) * i1S;
  float a2S = -(g1S - c2s[brk]) * i2S;
  float a3S = (nsS - g2S - g3S + c3s[brk]) * i1S * i2S;

  float hv[3] = { huber1(a1S - a1T), huber1(a2S - a2T), huber1(a3S - a3T) };
  for (int q = 0; q < 3; ++q) {
    red[threadIdx.x] = hv[q]; __syncthreads();
    for (int o = 128; o > 0; o >>= 1) { if (threadIdx.x < o) red[threadIdx.x] += red[threadIdx.x + o]; __syncthreads(); }
    if (threadIdx.x == 0) atomicAdd(&acc[q], red[0]);
    __syncthreads();
  }
}

__global__ void k_final(const float* __restrict__ acc, float* __restrict__ out) {
  float total = (float)((long)B_ * R_ * NS_ * K_ * 3);
  out[0] = (acc[0] + acc[1] + acc[2]) / total;
}

extern "C" void kernel_launch(void* const* d_in, const int* in_sizes, int n_in,
                              void* d_out, int out_size, void* d_ws, size_t ws_size,
                              hipStream_t stream) {
  (void)in_sizes; (void)n_in; (void)out_size; (void)ws_size;
  const float* T    = (const float*)d_in[0];
  const float* S    = (const float*)d_in[1];
  const int*   refp = (const int*)d_in[2];
  const int*   shp  = (const int*)d_in[3];
  float* out = (float*)d_out;

  float* ws = (float*)d_ws;
  size_t o = 0;
  auto alloc = [&](size_t n) { float* p = ws + o; o += n; return p; };
  float* ref_t = alloc((size_t)B_ * R_ * D_);
  float* ref_s = alloc((size_t)B_ * R_ * D_);
  float* simh  = alloc((size_t)B_ * R_ * K_ * D_);
  float* b1_t  = alloc((size_t)B_ * R_ * K_ * D_);
  float* b1_s  = alloc((size_t)B_ * R_ * K_ * D_);
  float* inv_extra = alloc((size_t)B_ * M_);
  float* sim   = alloc((size_t)B_ * R_ * M_);
  float* sh_t  = alloc((size_t)B_ * NS_ * D_);
  float* sh_s  = alloc((size_t)B_ * NS_ * D_);
  float* G1t   = alloc((size_t)B_ * NS_ * RK_);
  float* G1s   = alloc((size_t)B_ * NS_ * RK_);
  float* G2t   = alloc((size_t)B_ * NS_ * RK_);
  float* G2s   = alloc((size_t)B_ * NS_ * RK_);
  float* G3t   = alloc((size_t)B_ * NS_ * R_);
  float* G3s   = alloc((size_t)B_ * NS_ * R_);
  float* nr_t  = alloc(B_ * R_);
  float* nr_s  = alloc(B_ * R_);
  float* nh    = alloc(B_ * R_ * K_);
  float* c1t = alloc(B_ * R_ * K_); float* c2t = alloc(B_ * R_ * K_); float* c3t = alloc(B_ * R_ * K_);
  float* c1s = alloc(B_ * R_ * K_); float* c2s = alloc(B_ * R_ * K_); float* c3s = alloc(B_ * R_ * K_);
  float* ns_t = alloc(B_ * NS_);
  float* ns_s = alloc(B_ * NS_);
  float* acc  = alloc(4);
  int*   idx  = (int*)alloc(B_ * R_ * K_);

  k_init<<<1, 32, 0, stream>>>(acc);
  k_gather_ref<<<B_ * R_, 256, 0, stream>>>(T, S, refp, ref_t, ref_s, nr_t, nr_s);
  k_inv_extra<<<B_ * M_, 256, 0, stream>>>(T, inv_extra);

  {
    int nTiles = (M_ + 15) / 16;                 // 343
    dim3 g((nTiles + 3) / 4, R_ / 16, B_);       // 86 x 8 x 2, 128 thr (4 waves)
    k_gemm_rowdot<<<g, 128, 0, stream>>>(ref_t, T, sim, inv_extra,
        R_, M_, D_, (long)R_ * D_, (long)ST_ * P_ * D_, (long)R_ * M_, 1);
  }
  k_topk<<<B_ * R_, 32, 0, stream>>>(sim, idx);
  k_brk<<<B_ * R_ * K_, 256, 0, stream>>>(T, idx, ref_t, ref_s, simh, b1_t, b1_s,
                                          nh, c1t, c2t, c3t, c1s, c2s, c3s);

  dim3 g512(RK_ / 16 / 4, NS_ / 16, B_);   // 8 x 8 x 2
  dim3 g128(R_ / 16 / 4, NS_ / 16, B_);    // 2 x 8 x 2
  for (int f = 1; f <= 3; ++f) {
    int tf = 2 * f;                        // STUDENT_FRAMES[1:] = 2,4,6
    k_gather_sh<<<B_ * NS_, 256, 0, stream>>>(T, S, shp, tf, f, sh_t, sh_s, ns_t, ns_s);
    k_gemm_rowdot<<<g512, 128, 0, stream>>>(sh_t, b1_t, G1t, nullptr,
        NS_, RK_, D_, (long)NS_ * D_, (long)RK_ * D_, (long)NS_ * RK_, 0);
    k_gemm_rowdot<<<g512, 128, 0, stream>>>(sh_s, b1_s, G1s, nullptr,
        NS_, RK_, D_, (long)NS_ * D_, (long)RK_ * D_, (long)NS_ * RK_, 0);
    k_gemm_rowdot<<<g512, 128, 0, stream>>>(sh_t, simh, G2t, nullptr,
        NS_, RK_, D_, (long)NS_ * D_, (long)RK_ * D_, (long)NS_ * RK_, 0);
    k_gemm_rowdot<<<g512, 128, 0, stream>>>(sh_s, simh, G2s, nullptr,
        NS_, RK_, D_, (long)NS_ * D_, (long)RK_ * D_, (long)NS_ * RK_, 0);
    k_gemm_rowdot<<<g128, 128, 0, stream>>>(sh_t, ref_t, G3t, nullptr,
        NS_, R_, D_, (long)NS_ * D_, (long)R_ * D_, (long)NS_ * R_, 0);
    k_gemm_rowdot<<<g128, 128, 0, stream>>>(sh_s, ref_s, G3s, nullptr,
        NS_, R_, D_, (long)NS_ * D_, (long)R_ * D_, (long)NS_ * R_, 0);
    k_epilogue<<<(B_ * R_ * NS_ * K_) / 256, 256, 0, stream>>>(
        G1t, G1s, G2t, G2s, G3t, G3s, ns_t, ns_s, nr_t, nr_s, nh,
        c1t, c2t, c3t, c1s, c2s, c3s, acc);
  }
  k_final<<<1, 1, 0, stream>>>(acc, out);
}